// GNNRecommendationModel_27736898798028
// MI455X (gfx1250) — compile-verified
//
#include <hip/hip_runtime.h>
#include <hip/hip_bf16.h>

#define NN 50000
#define NE 800000

typedef __attribute__((ext_vector_type(2))) float v2f;
typedef __attribute__((ext_vector_type(8))) float v8f;

// ---------------- utility kernels ----------------

__global__ void zero_f4(float4* __restrict__ p, int n4) {
    int i = blockIdx.x * blockDim.x + threadIdx.x;
    if (i < n4) p[i] = make_float4(0.f, 0.f, 0.f, 0.f);
}

__global__ void count_edges(const int* __restrict__ dst, float* __restrict__ cnt, int E) {
    int e = blockIdx.x * blockDim.x + threadIdx.x;
    if (e < E) unsafeAtomicAdd(&cnt[dst[e]], 1.0f);   // hardware global_atomic_add_f32
}

// scatter-add of source features into per-destination sums: agg[dst] += X[src]
// one thread = one edge x 4 channels (float4 gather, 4 fp32 atomics)
template <int C>
__global__ void scatter_sum(const float* __restrict__ X,
                            const int* __restrict__ src,
                            const int* __restrict__ dst,
                            float* __restrict__ agg, int E) {
    constexpr int P = C / 4;
    int idx = blockIdx.x * blockDim.x + threadIdx.x;
    if (idx >= E * P) return;
    int e  = idx / P;
    int c4 = (idx - e * P) * 4;
    int s = src[e], d = dst[e];
    const float4 v = *reinterpret_cast<const float4*>(X + (size_t)s * C + c4);
    float* b = agg + (size_t)d * C + c4;
    unsafeAtomicAdd(b + 0, v.x);
    unsafeAtomicAdd(b + 1, v.y);
    unsafeAtomicAdd(b + 2, v.z);
    unsafeAtomicAdd(b + 3, v.w);
}

// Fused SAGEConv GEMM: out = (agg/max(cnt,1)) @ Wl^T + X @ Wr^T + bl  (+ReLU)
// One block = one 16-row node tile; wave w computes output-channel tile [16w,16w+16).
// fp32 matrix cores: V_WMMA_F32_16X16X4_F32, K stepped by 4, A tile staged in LDS
// via b128 loads/stores (rows are contiguous from m0*CIN).
template <int CIN, int COUT, bool RELU>
__global__ __launch_bounds__(2 * COUT) void sage_gemm(
    const float* __restrict__ agg, const float* __restrict__ cnt,
    const float* __restrict__ X,   const float* __restrict__ Wl,
    const float* __restrict__ bl,  const float* __restrict__ Wr,
    float* __restrict__ out) {
    constexpr int NW  = COUT / 16;            // waves per block
    constexpr int SH4 = (CIN == 128) ? 5 : 4; // log2(CIN/4): row from float4 index
    constexpr int Q   = 16 * CIN / 4;         // float4 elements in one A tile
    __shared__ float tileA[16 * CIN];
    __shared__ float invs[16];

    const int tid  = threadIdx.x;
    const int lane = tid & 31;
    const int wv   = tid >> 5;
    const int m0   = blockIdx.x * 16;

    if (tid < 16) {
        float c = cnt[m0 + tid];
        invs[tid] = 1.0f / fmaxf(c, 1.0f);
    }
    __syncthreads();

    // stage A = mean tile (scale folded in), vectorized b128
    {
        const float4* src4 = reinterpret_cast<const float4*>(agg + (size_t)m0 * CIN);
        float4* dst4 = reinterpret_cast<float4*>(tileA);
#pragma unroll
        for (int i = tid; i < Q; i += 32 * NW) {
            float4 v = src4[i];
            float s = invs[i >> SH4];
            v.x *= s; v.y *= s; v.z *= s; v.w *= s;
            dst4[i] = v;
        }
    }
    __syncthreads();

    const int mn   = lane & 15;          // A row index == B col index
    const int kOff = (lane >> 4) * 2;    // lanes 16-31 hold K+2,K+3
    const int n0   = wv * 16;

    v8f acc = {};
    const float* wlrow = Wl + (size_t)(n0 + mn) * CIN;
#pragma unroll 4
    for (int k = 0; k < CIN; k += 4) {
        v2f a = *reinterpret_cast<const v2f*>(&tileA[mn * CIN + k + kOff]);
        v2f b = *reinterpret_cast<const v2f*>(&wlrow[k + kOff]);
        acc = __builtin_amdgcn_wmma_f32_16x16x4_f32(false, a, false, b,
                                                    (short)0, acc, false, false);
    }
    __syncthreads();

    // restage A = self-feature tile, vectorized b128
    {
        const float4* src4 = reinterpret_cast<const float4*>(X + (size_t)m0 * CIN);
        float4* dst4 = reinterpret_cast<float4*>(tileA);
#pragma unroll
        for (int i = tid; i < Q; i += 32 * NW) dst4[i] = src4[i];
    }
    __syncthreads();

    const float* wrrow = Wr + (size_t)(n0 + mn) * CIN;
#pragma unroll 4
    for (int k = 0; k < CIN; k += 4) {
        v2f a = *reinterpret_cast<const v2f*>(&tileA[mn * CIN + k + kOff]);
        v2f b = *reinterpret_cast<const v2f*>(&wrrow[k + kOff]);
        acc = __builtin_amdgcn_wmma_f32_16x16x4_f32(false, a, false, b,
                                                    (short)0, acc, false, false);
    }

    const float bias = bl[n0 + mn];
#pragma unroll
    for (int v = 0; v < 8; ++v) {
        int row = v + ((lane >> 4) << 3);      // C/D layout: VGPR v -> row v / v+8
        float val = acc[v] + bias;
        if (RELU) val = fmaxf(val, 0.0f);
        out[(size_t)(m0 + row) * COUT + n0 + mn] = val;
    }
}

// row-wise log_softmax over 64 channels, one wave32 per row (2 cols/lane)
__global__ void logsoftmax64(const float* __restrict__ in, float* __restrict__ out, int nrows) {
    int lane = threadIdx.x & 31;
    int wv   = threadIdx.x >> 5;
    int row  = blockIdx.x * 8 + wv;
    if (row >= nrows) return;
    float2 v = *reinterpret_cast<const float2*>(in + (size_t)row * 64 + lane * 2);
    float m = fmaxf(v.x, v.y);
    for (int o = 16; o > 0; o >>= 1) m = fmaxf(m, __shfl_xor(m, o, 32));
    float s = expf(v.x - m) + expf(v.y - m);
    for (int o = 16; o > 0; o >>= 1) s += __shfl_xor(s, o, 32);
    float ls = m + logf(s);
    float2 r; r.x = v.x - ls; r.y = v.y - ls;
    *reinterpret_cast<float2*>(out + (size_t)row * 64 + lane * 2) = r;
}

// ---------------- launcher ----------------

static inline size_t al256(size_t v) { return (v + 255) & ~(size_t)255; }

extern "C" void kernel_launch(void* const* d_in, const int* in_sizes, int n_in,
                              void* d_out, int out_size, void* d_ws, size_t ws_size,
                              hipStream_t stream) {
    const float* x   = (const float*)d_in[0];
    const int*   ei  = (const int*)d_in[1];      // edge_index flattened [2, E]
    const float* Wl1 = (const float*)d_in[2];
    const float* bl1 = (const float*)d_in[3];
    const float* Wr1 = (const float*)d_in[4];
    const float* Wl2 = (const float*)d_in[5];
    const float* bl2 = (const float*)d_in[6];
    const float* Wr2 = (const float*)d_in[7];
    const float* Wl3 = (const float*)d_in[8];
    const float* bl3 = (const float*)d_in[9];
    const float* Wr3 = (const float*)d_in[10];
    const float* Wl4 = (const float*)d_in[11];
    const float* bl4 = (const float*)d_in[12];
    const float* Wr4 = (const float*)d_in[13];
    float* out = (float*)d_out;

    const int* src = ei;
    const int* dst = ei + NE;

    char* w = (char*)d_ws;
    size_t off = 0;
    float* cnt = (float*)(w + off); off += al256((size_t)NN * 4);
    float* agg = (float*)(w + off); off += al256((size_t)NN * 128 * 4);
    float* hA  = (float*)(w + off); off += al256((size_t)NN * 128 * 4);
    float* hB  = (float*)(w + off);

    const int TB = 256;
    auto zero = [&](float* p, size_t n) {
        int n4 = (int)(n / 4);
        zero_f4<<<(n4 + TB - 1) / TB, TB, 0, stream>>>((float4*)p, n4);
    };

    // degree counts (graph is constant across layers)
    zero(cnt, NN);
    count_edges<<<(NE + TB - 1) / TB, TB, 0, stream>>>(dst, cnt, NE);

    // layer 1: x(128) -> hA(128), ReLU
    zero(agg, (size_t)NN * 128);
    scatter_sum<128><<<(NE * 32 + TB - 1) / TB, TB, 0, stream>>>(x, src, dst, agg, NE);
    sage_gemm<128, 128, true><<<NN / 16, 256, 0, stream>>>(agg, cnt, x, Wl1, bl1, Wr1, hA);

    // layer 2: hA(128) -> hB(128), ReLU
    zero(agg, (size_t)NN * 128);
    scatter_sum<128><<<(NE * 32 + TB - 1) / TB, TB, 0, stream>>>(hA, src, dst, agg, NE);
    sage_gemm<128, 128, true><<<NN / 16, 256, 0, stream>>>(agg, cnt, hA, Wl2, bl2, Wr2, hB);

    // layer 3: hB(128) -> hA(64), ReLU
    zero(agg, (size_t)NN * 128);
    scatter_sum<128><<<(NE * 32 + TB - 1) / TB, TB, 0, stream>>>(hB, src, dst, agg, NE);
    sage_gemm<128, 64, true><<<NN / 16, 128, 0, stream>>>(agg, cnt, hB, Wl3, bl3, Wr3, hA);

    // layer 4: hA(64) -> hB(64), no ReLU
    zero(agg, (size_t)NN * 64);
    scatter_sum<64><<<(NE * 16 + TB - 1) / TB, TB, 0, stream>>>(hA, src, dst, agg, NE);
    sage_gemm<64, 64, false><<<NN / 16, 128, 0, stream>>>(agg, cnt, hA, Wl4, bl4, Wr4, hB);

    // log_softmax over 64 channels
    logsoftmax64<<<NN / 8, 256, 0, stream>>>(hB, out, NN);
}